// Model_28432683499856
// MI455X (gfx1250) — compile-verified
//
#include <hip/hip_runtime.h>
#include <math.h>

// ---------------------------------------------------------------------------
// Problem constants
// ---------------------------------------------------------------------------
#define NN     128            // nodes
#define NF     32768          // frames
#define CHL    128            // chunk length (steps)
#define NCH    256            // number of chunks (NCH*CHL == NF)
#define NS     524288         // output samples
#define NSPEC  16385          // rfft bins of 32768-point signal

typedef __attribute__((ext_vector_type(2))) float v2f;
typedef __attribute__((ext_vector_type(8))) float v8f;

// workspace layout (float offsets)
#define OFF_AHAT   0u                      // 128x128  A/m (row-scaled)
#define OFF_M      16384u                  // 256x256  state transition M
#define OFF_PA     81920u                  // 256x256  squaring ping
#define OFF_PB     147456u                 // 256x256  squaring pong
#define OFF_Q      212992u                 // 128      q = A^T mix
#define OFF_W2     213120u                 // 128      mix*m
#define OFF_CMK    213248u                 // 128      constrained mask {0,1}
#define OFF_DG     213376u                 // 256x256  chunk zero-init end states
#define OFF_BND    278912u                 // 256x256  chunk boundary states s_{gL}
#define OFF_V      344448u                 // 128x256  V_tau = (M^T)^tau qtilde
#define OFF_REC    377216u                 // 32768    rec signal
#define OFF_CSP    409984u                 // 16385x2  spectrum C_k
#define OFF_U0     442880u                 // 32768x128 zero-init u trajectories (16 MB)
#define OFF_R2     4637184u                // 128x256  R2[tau][g] = V_tau . s_g
// total ~ 4,669,952 floats ~ 18.7 MB of d_ws

// ---------------------------------------------------------------------------
// k_prep: build m_eff, softmax(mix), A, Ahat=A/m, q=A^T mix, w2=mix*m, M(256x256)
// 1 block x 256 threads, dyn LDS = (16384+384)*4 B
// ---------------------------------------------------------------------------
__global__ void k_prep(const float* __restrict__ masses,
                       const float* __restrict__ tensions,
                       const float* __restrict__ mixer,
                       const float* __restrict__ cmask_in,
                       float* __restrict__ Ahat, float* __restrict__ M,
                       float* __restrict__ q, float* __restrict__ w2,
                       float* __restrict__ cmk) {
  extern __shared__ float smem_f[];
  float* A_sh   = smem_f;            // 16384
  float* m_sh   = smem_f + 16384;    // 128
  float* mix_sh = smem_f + 16512;    // 128
  float* c_sh   = smem_f + 16640;    // 128
  const int tid = threadIdx.x;

  if (tid < NN) {
    m_sh[tid]   = fabsf(masses[tid]) * 100.0f + 1e-8f;
    c_sh[tid]   = (cmask_in[tid] > 0.0f) ? 1.0f : 0.0f;   // straight-through hard mask
    mix_sh[tid] = mixer[tid];
  }
  __syncthreads();
  if (tid == 0) {                                          // softmax (tiny, serial)
    float mx = mix_sh[0];
    for (int i = 1; i < NN; ++i) mx = fmaxf(mx, mix_sh[i]);
    float s = 0.0f;
    for (int i = 0; i < NN; ++i) { float e = expf(mix_sh[i] - mx); mix_sh[i] = e; s += e; }
    float inv = 1.0f / s;
    for (int i = 0; i < NN; ++i) mix_sh[i] *= inv;
  }
  __syncthreads();
  // off-diagonal of A: +-z, z = 4*|T_sym| + 1e-8 (conn==1 everywhere since z>0)
  for (int idx = tid; idx < NN * NN; idx += 256) {
    int r = idx >> 7, c = idx & 127;
    if (r != c) {
      float tv = (r < c) ? tensions[r * NN + c] : tensions[c * NN + r];
      float z  = fabsf(tv) * 4.0f + 1e-8f;
      A_sh[idx] = (c < r) ? -z : z;
    } else A_sh[idx] = 0.0f;
  }
  __syncthreads();
  // diagonal: A[k][k] = sum_{i<k} z - sum_{i>k} z = -(row sum of off-diag)
  if (tid < NN) {
    float s = 0.0f;
    for (int c = 0; c < NN; ++c) s += A_sh[tid * NN + c];
    A_sh[tid * NN + tid] = -s;
  }
  __syncthreads();
  if (tid < NN) {
    float s = 0.0f;
    for (int k = 0; k < NN; ++k) s = fmaf(mix_sh[k], A_sh[k * NN + tid], s);
    q[tid]  = s;                       // rec_t = q . u_t + w2 . ftx_t
    w2[tid] = mix_sh[tid] * m_sh[tid];
    cmk[tid] = c_sh[tid];
  }
  for (int idx = tid; idx < NN * NN; idx += 256)
    Ahat[idx] = A_sh[idx] / m_sh[idx >> 7];
  // M = [[I + C*Ahat, C], [0.95*Ahat, 0.95*I]] on state s=(u,vel)
  for (int idx = tid; idx < 256 * 256; idx += 256) {
    int r = idx >> 8, c = idx & 255;
    float v;
    if (r < NN) {
      if (c < NN) v = ((r == c) ? 1.0f : 0.0f) + c_sh[r] * (A_sh[r * NN + c] / m_sh[r]);
      else        v = ((c - NN) == r) ? c_sh[r] : 0.0f;
    } else {
      int rr = r - NN;
      if (c < NN) v = 0.95f * (A_sh[rr * NN + c] / m_sh[rr]);
      else        v = ((c - NN) == rr) ? 0.95f : 0.0f;
    }
    M[idx] = v;
  }
}

// ---------------------------------------------------------------------------
// k_matsq: D = S*S for 256x256 f32 via V_WMMA_F32_16X16X4_F32.
// grid 32 x 256 threads (8 waves/WG, one 16x16 output tile per wave, K=256).
// ---------------------------------------------------------------------------
__global__ void k_matsq(const float* __restrict__ S, float* __restrict__ D) {
  const int wave = threadIdx.x >> 5, lane = threadIdx.x & 31;
  const int tile = blockIdx.x * 8 + wave;          // 0..255
  const int tr = tile >> 4, tc = tile & 15;
  const int l16 = lane & 15, hf = lane >> 4;
  v8f c = {0.f, 0.f, 0.f, 0.f, 0.f, 0.f, 0.f, 0.f};
  for (int k = 0; k < 256; k += 4) {
    const int ka = k + 2 * hf;                     // K layout: lanes16-31 hold K+2,K+3
    v2f a, b;
    a.x = S[(tr * 16 + l16) * 256 + ka];           // A 16x4 fragment
    a.y = S[(tr * 16 + l16) * 256 + ka + 1];
    b.x = S[ka * 256 + tc * 16 + l16];             // B 4x16 fragment
    b.y = S[(ka + 1) * 256 + tc * 16 + l16];
    c = __builtin_amdgcn_wmma_f32_16x16x4_f32(false, a, false, b, (short)0, c,
                                              false, false);
  }
  #pragma unroll
  for (int r = 0; r < 8; ++r)                      // C/D: vgpr r -> row r (+8 for hi half)
    D[(tr * 16 + hf * 8 + r) * 256 + tc * 16 + l16] = c[r];
}

// ---------------------------------------------------------------------------
// k_corr: R2 = V (128x256) x bnd^T (256x256) via WMMA; R2[tau][g] = V_tau . s_g.
// bnd is (g,i) row-major, so B'[k][n] = bnd[n*256+k]: b.x/b.y are consecutive.
// grid 16 x 256 threads (8 waves/WG, 128 output tiles, K=256).
// ---------------------------------------------------------------------------
__global__ void k_corr(const float* __restrict__ V, const float* __restrict__ bnd,
                       float* __restrict__ R2) {
  const int wave = threadIdx.x >> 5, lane = threadIdx.x & 31;
  const int tile = blockIdx.x * 8 + wave;          // 0..127
  const int tr = tile >> 4, tc = tile & 15;        // tr: tau-tile (0..7), tc: g-tile (0..15)
  const int l16 = lane & 15, hf = lane >> 4;
  v8f c = {0.f, 0.f, 0.f, 0.f, 0.f, 0.f, 0.f, 0.f};
  for (int k = 0; k < 256; k += 4) {
    const int ka = k + 2 * hf;
    v2f a, b;
    a.x = V[(tr * 16 + l16) * 256 + ka];
    a.y = V[(tr * 16 + l16) * 256 + ka + 1];
    b.x = bnd[(tc * 16 + l16) * 256 + ka];         // B'[ka][n] = bnd[n][ka]
    b.y = bnd[(tc * 16 + l16) * 256 + ka + 1];
    c = __builtin_amdgcn_wmma_f32_16x16x4_f32(false, a, false, b, (short)0, c,
                                              false, false);
  }
  #pragma unroll
  for (int r = 0; r < 8; ++r)
    R2[(tr * 16 + hf * 8 + r) * 256 + tc * 16 + l16] = c[r];
}

// ---------------------------------------------------------------------------
// k_sim: zero-init chunk simulations. grid NCH x 128 threads (lane == node).
// A-row in 128 VGPRs; u broadcast via LDS float4 loads (conflict-free).
// Stores u^0_t trajectory (for rec) and end state d_g.
// ---------------------------------------------------------------------------
__global__ void k_sim(const float* __restrict__ Ahat, const float* __restrict__ cmk,
                      const float* __restrict__ forces,
                      float* __restrict__ u0buf, float* __restrict__ dg) {
  __shared__ __align__(16) float u_sh[NN];
  const int k = threadIdx.x;                       // node id
  const int g = blockIdx.x;                        // chunk id
  float rowr[NN];
  #pragma unroll
  for (int i = 0; i < NN; ++i) rowr[i] = Ahat[k * NN + i];
  const float cm = cmk[k];
  float u = 0.0f, vel = 0.0f;
  u_sh[k] = 0.0f;
  __syncthreads();
  const float4* u4 = (const float4*)u_sh;
  for (int tau = 0; tau < CHL; ++tau) {
    const int t = g * CHL + tau;
    u0buf[t * NN + k] = u;                         // record u_t (pre-update)
    float acc = ((tau & 15) == 0) ? forces[(t >> 4) * (NN * 3) + 3 * k] : 0.0f;
    #pragma unroll
    for (int i = 0; i < NN / 4; ++i) {
      float4 uu = u4[i];
      acc = fmaf(rowr[4 * i + 0], uu.x, acc);
      acc = fmaf(rowr[4 * i + 1], uu.y, acc);
      acc = fmaf(rowr[4 * i + 2], uu.z, acc);
      acc = fmaf(rowr[4 * i + 3], uu.w, acc);
    }
    vel += acc;                                    // vel' = vel + acc
    u   = fmaf(vel, cm, u);                        // u += c * vel'
    vel *= 0.95f;                                  // damping
    __syncthreads();                               // all reads of old u done
    u_sh[k] = u;
    __syncthreads();
  }
  dg[g * 256 + k]      = u;                        // d_g = zero-init chunk response
  dg[g * 256 + NN + k] = vel;
}

// ---------------------------------------------------------------------------
// k_vcomp: V_tau = (M^T)^tau * qtilde, tau = 0..CHL-1. 1 block x 256 threads.
// dyn LDS = (65536+256)*4 B. M row-major -> column dot is bank-conflict-free.
// ---------------------------------------------------------------------------
__global__ void k_vcomp(const float* __restrict__ Mmat, const float* __restrict__ q,
                        float* __restrict__ Vbuf) {
  extern __shared__ float smem_f[];
  float* Msh = smem_f;              // 65536
  float* vsh = smem_f + 65536;      // 256
  const int tid = threadIdx.x;
  for (int i = tid; i < 256 * 256; i += 256) Msh[i] = Mmat[i];
  vsh[tid] = (tid < NN) ? q[tid] : 0.0f;
  __syncthreads();
  for (int tau = 0; tau < CHL; ++tau) {
    Vbuf[tau * 256 + tid] = vsh[tid];
    float acc = 0.0f;
    for (int i = 0; i < 256; ++i) acc = fmaf(Msh[i * 256 + tid], vsh[i], acc);
    __syncthreads();
    vsh[tid] = acc;
    __syncthreads();
  }
}

// ---------------------------------------------------------------------------
// k_bound: boundary scan s_{g+1} = M^CHL * s_g + d_g, s_0 = 0.
// 1 block x 256 threads. P stored transposed in LDS for conflict-free matvec.
// ---------------------------------------------------------------------------
__global__ void k_bound(const float* __restrict__ P, const float* __restrict__ dg,
                        float* __restrict__ bnd) {
  extern __shared__ float smem_f[];
  float* Q    = smem_f;             // Q[i*256+j] = P[j*256+i]
  float* s_sh = smem_f + 65536;
  const int tid = threadIdx.x;
  for (int idx = tid; idx < 256 * 256; idx += 256) {
    int r = idx >> 8, c = idx & 255;
    Q[c * 256 + r] = P[idx];
  }
  s_sh[tid] = 0.0f;
  __syncthreads();
  for (int g = 0; g < NCH; ++g) {
    bnd[g * 256 + tid] = s_sh[tid];
    float acc = dg[g * 256 + tid];
    for (int i = 0; i < 256; ++i) acc = fmaf(Q[i * 256 + tid], s_sh[i], acc);
    __syncthreads();
    s_sh[tid] = acc;
    __syncthreads();
  }
}

// ---------------------------------------------------------------------------
// k_rec: rec_t = q.u0_t + R2[tau][g] + (t%16==0 ? w2 . forces_x : 0)
// grid 4096 x 256 (one wave per t, wave32 shuffle reduction).
// ---------------------------------------------------------------------------
__global__ void k_rec(const float* __restrict__ q, const float* __restrict__ w2,
                      const float* __restrict__ R2, const float* __restrict__ u0,
                      const float* __restrict__ forces, float* __restrict__ rec) {
  const int wave = threadIdx.x >> 5, lane = threadIdx.x & 31;
  const int t = blockIdx.x * 8 + wave;
  const int g = t >> 7, tau = t & (CHL - 1);
  float p = 0.0f;
  for (int i = lane; i < NN; i += 32)  p = fmaf(q[i], u0[t * NN + i], p);
  if ((t & 15) == 0) {
    const int r = t >> 4;
    for (int i = lane; i < NN; i += 32) p = fmaf(w2[i], forces[r * (NN * 3) + 3 * i], p);
  }
  for (int off = 16; off > 0; off >>= 1) p += __shfl_down(p, off);
  if (lane == 0) rec[t] = p + R2[tau * 256 + g];
}

// ---------------------------------------------------------------------------
// k_fft_fwd: 32768-pt complex FFT (imag=0) in LDS (256 KB), emit C_k, k<=16384.
// 1 block x 1024 threads. Radix-2 DIT, bit-reversed load, e^{-} twiddles.
// ---------------------------------------------------------------------------
__global__ void k_fft_fwd(const float* __restrict__ rec, float* __restrict__ Cout) {
  extern __shared__ float smem_f[];
  float2* X = (float2*)smem_f;
  const int tid = threadIdx.x;
  for (int i = tid; i < NF; i += 1024) {
    int r = __brev((unsigned)i) >> 17;             // 15-bit reversal
    X[i] = make_float2(rec[r], 0.0f);
  }
  __syncthreads();
  for (int s = 1; s <= 15; ++s) {
    const int m = 1 << s, h = m >> 1;
    for (int b = tid; b < NF / 2; b += 1024) {
      const int pos = b & (h - 1);
      const int i0 = ((b >> (s - 1)) << s) + pos, i1 = i0 + h;
      float sn, cs;
      __sincosf(-6.283185307179586f * (float)pos / (float)m, &sn, &cs);
      float2 x0 = X[i0], x1 = X[i1];
      float tr = cs * x1.x - sn * x1.y, ti = cs * x1.y + sn * x1.x;
      X[i0] = make_float2(x0.x + tr, x0.y + ti);
      X[i1] = make_float2(x0.x - tr, x0.y - ti);
    }
    __syncthreads();
  }
  for (int k = tid; k < NSPEC; k += 1024) {
    Cout[2 * k]     = X[k].x;
    Cout[2 * k + 1] = X[k].y;
  }
}

// ---------------------------------------------------------------------------
// k_fft_inv: out[16t+s] = (1/32768) Re IDFT_32768(G_k e^{2pi i k s/524288})[t]
// with G_0 = C_0, G_k = 2 C_k (k<=16384), else 0. grid 16 (one phase s per WG).
// ---------------------------------------------------------------------------
__global__ void k_fft_inv(const float* __restrict__ Cin, float* __restrict__ out) {
  extern __shared__ float smem_f[];
  float2* X = (float2*)smem_f;
  const int tid = threadIdx.x, sph = blockIdx.x;
  const float W0 = 1.1984224905082319e-05f;        // 2*pi/524288
  for (int i = tid; i < NF; i += 1024) {
    const int k = __brev((unsigned)i) >> 17;
    float2 h;
    if (k == 0) {
      h = make_float2(Cin[0], Cin[1]);
    } else if (k <= 16384) {
      float cr = Cin[2 * k], ci = Cin[2 * k + 1];
      float sn, cs;
      __sincosf(W0 * (float)(k * sph), &sn, &cs);
      h = make_float2(2.0f * (cr * cs - ci * sn), 2.0f * (cr * sn + ci * cs));
    } else {
      h = make_float2(0.0f, 0.0f);
    }
    X[i] = h;
  }
  __syncthreads();
  for (int s = 1; s <= 15; ++s) {
    const int m = 1 << s, h2 = m >> 1;
    for (int b = tid; b < NF / 2; b += 1024) {
      const int pos = b & (h2 - 1);
      const int i0 = ((b >> (s - 1)) << s) + pos, i1 = i0 + h2;
      float sn, cs;
      __sincosf(6.283185307179586f * (float)pos / (float)m, &sn, &cs);  // inverse: e^{+}
      float2 x0 = X[i0], x1 = X[i1];
      float tr = cs * x1.x - sn * x1.y, ti = cs * x1.y + sn * x1.x;
      X[i0] = make_float2(x0.x + tr, x0.y + ti);
      X[i1] = make_float2(x0.x - tr, x0.y - ti);
    }
    __syncthreads();
  }
  const float scale = 1.0f / 32768.0f;
  for (int t = tid; t < NF; t += 1024)
    out[(t << 4) + sph] = X[t].x * scale;
}

// ---------------------------------------------------------------------------
// host driver
// ---------------------------------------------------------------------------
extern "C" void kernel_launch(void* const* d_in, const int* in_sizes, int n_in,
                              void* d_out, int out_size, void* d_ws, size_t ws_size,
                              hipStream_t stream) {
  // inputs: nodes(unused), masses, tensions, mixer, forces, constrained_mask
  const float* masses   = (const float*)d_in[1];
  const float* tensions = (const float*)d_in[2];
  const float* mixer    = (const float*)d_in[3];
  const float* forces   = (const float*)d_in[4];
  const float* cmask    = (const float*)d_in[5];
  float* out = (float*)d_out;
  float* ws  = (float*)d_ws;

  float* Ahat = ws + OFF_AHAT;
  float* M    = ws + OFF_M;
  float* PA   = ws + OFF_PA;
  float* PB   = ws + OFF_PB;
  float* q    = ws + OFF_Q;
  float* w2   = ws + OFF_W2;
  float* cmk  = ws + OFF_CMK;
  float* dg   = ws + OFF_DG;
  float* bnd  = ws + OFF_BND;
  float* Vbuf = ws + OFF_V;
  float* rec  = ws + OFF_REC;
  float* Csp  = ws + OFF_CSP;
  float* u0   = ws + OFF_U0;
  float* R2   = ws + OFF_R2;

  const size_t lds_prep = (16384 + 384) * sizeof(float);
  const size_t lds_mat  = (65536 + 256) * sizeof(float);
  const size_t lds_fft  = NF * 2 * sizeof(float);

  k_prep<<<1, 256, lds_prep, stream>>>(masses, tensions, mixer, cmask,
                                       Ahat, M, q, w2, cmk);
  // M^128 by 7 WMMA squarings: M->PA->PB->...->PA
  k_matsq<<<32, 256, 0, stream>>>(M,  PA);
  k_matsq<<<32, 256, 0, stream>>>(PA, PB);
  k_matsq<<<32, 256, 0, stream>>>(PB, PA);
  k_matsq<<<32, 256, 0, stream>>>(PA, PB);
  k_matsq<<<32, 256, 0, stream>>>(PB, PA);
  k_matsq<<<32, 256, 0, stream>>>(PA, PB);
  k_matsq<<<32, 256, 0, stream>>>(PB, PA);          // PA = M^128

  k_vcomp<<<1, 256, lds_mat, stream>>>(M, q, Vbuf);
  k_sim<<<NCH, NN, 0, stream>>>(Ahat, cmk, forces, u0, dg);
  k_bound<<<1, 256, lds_mat, stream>>>(PA, dg, bnd);
  k_corr<<<16, 256, 0, stream>>>(Vbuf, bnd, R2);    // WMMA: R2 = V * bnd^T
  k_rec<<<NF / 8, 256, 0, stream>>>(q, w2, R2, u0, forces, rec);
  k_fft_fwd<<<1, 1024, lds_fft, stream>>>(rec, Csp);
  k_fft_inv<<<16, 1024, lds_fft, stream>>>(Csp, out);
}